// RetinaUNetCore_14920716387180
// MI455X (gfx1250) — compile-verified
//
#include <hip/hip_runtime.h>
#include <hip/hip_bf16.h>
#include <stdint.h>

// ---------------- problem constants (from reference) ----------------
#define BATCH      8
#define N_CLASSES  3
#define N_FG       2
#define N_ANCHORS  65280
#define N_PROP     (BATCH * N_ANCHORS)     // 522240
#define N_FLAT     (N_PROP * N_FG)         // 1044480
#define TOPK       6000
#define PAD_K      6016                    // 6000 padded to 16*376 tile rows
#define COL_WORDS  188                     // ceil(6000/32)
#define ROW_BLOCKS 375                     // 6000/16
#define COL_BLOCKS 188
#define CAND_MAX   8192
#define NBUCKETS   65536
#define SCALE_F    512.0f
#define GROUP_OFF  4096.0f
#define NMS_THRESH 0.5f

// output layout (floats): detections (6000x6) | class_ids (6000) | b (6000)
#define OUT_DET 0
#define OUT_CLS (TOPK * 6)
#define OUT_B   (TOPK * 7)

// ---------------- workspace layout (bytes) ----------------
#define HIST_OFF   0u
#define CNT_OFF    (HIST_OFF + NBUCKETS * 4u)        // 262144; 8 u32 counters
#define CAND_OFF   (CNT_OFF + 32u)                   // 262176 (8B aligned)
#define SCORES_OFF (CAND_OFF + CAND_MAX * 8u)        // 327712
#define FLAT_OFF   (SCORES_OFF + TOPK * 4u)          // 351712
#define SB_OFF     (FLAT_OFF + TOPK * 4u)            // 375712 (16B aligned)
#define AREAS_OFF  (SB_OFF + PAD_K * 16u)            // 471968
#define MASK_OFF   (AREAS_OFF + PAD_K * 4u)          // 496032 (~5.0 MB total)

typedef __attribute__((ext_vector_type(2))) float v2f;
typedef __attribute__((ext_vector_type(8))) float v8f;

#if __has_builtin(__builtin_amdgcn_wmma_f32_16x16x4_f32)
#define HAVE_WMMA_F32_4 1
#endif

static __device__ __forceinline__ unsigned ballot32(bool p) {
#if __has_builtin(__builtin_amdgcn_ballot_w32)
    return __builtin_amdgcn_ballot_w32(p);
#else
    return (unsigned)__ballot(p);
#endif
}

// ---------------- K1: zero histogram + counters ----------------
__global__ void init_kernel(unsigned* __restrict__ hist_and_cnt) {
    int i = blockIdx.x * blockDim.x + threadIdx.x;
    if (i < NBUCKETS + 8) hist_and_cnt[i] = 0u;
}

// ---------------- K2: 16-bit-key histogram of fg scores ----------------
// fg flat index f -> probs[(f>>1)*3 + 1 + (f&1)]; scores are positive floats
// so bits>>16 is order-preserving.
__global__ void hist_kernel(const float* __restrict__ probs,
                            unsigned* __restrict__ hist) {
    int f = blockIdx.x * blockDim.x + threadIdx.x;
    if (f >= N_FLAT) return;
    unsigned prop = (unsigned)f >> 1;
    __builtin_prefetch(probs + prop * 3 + 3072, 0, 0);   // global_prefetch_b8
    float s = probs[prop * 3 + 1 + (f & 1)];
    unsigned bits = __float_as_uint(s);
    atomicAdd(&hist[bits >> 16], 1u);
}

// ---------------- K3: find threshold bucket (suffix sum from top) ----------------
__global__ __launch_bounds__(256) void thresh_kernel(const unsigned* __restrict__ hist,
                                                     unsigned* __restrict__ cnt) {
    __shared__ unsigned csum[256];
    int tid = threadIdx.x;                 // each thread sums 256 buckets
    unsigned s = 0;
    for (int b = 0; b < 256; ++b) s += hist[tid * 256 + b];
    csum[tid] = s;
    __syncthreads();
    if (tid == 0) {
        unsigned cum = 0;
        int tb = 0;
        for (int c = 255; c >= 0; --c) {
            if (cum + csum[c] >= TOPK) {
                for (int b = 255; b >= 0; --b) {
                    cum += hist[c * 256 + b];
                    if (cum >= TOPK) { tb = c * 256 + b; break; }
                }
                cnt[1] = (unsigned)tb;
                return;
            }
            cum += csum[c];
        }
        cnt[1] = 0u;
    }
}

// ---------------- K4: compact candidates >= threshold bucket ----------------
// key = (score_bits << 32) | (~f)  -> descending sort == score desc, index asc.
__global__ void compact_kernel(const float* __restrict__ probs,
                               unsigned* __restrict__ cnt,
                               unsigned long long* __restrict__ cand) {
    int f = blockIdx.x * blockDim.x + threadIdx.x;
    if (f >= N_FLAT) return;
    unsigned T = cnt[1];
    unsigned prop = (unsigned)f >> 1;
    float s = probs[prop * 3 + 1 + (f & 1)];
    unsigned bits = __float_as_uint(s);
    if ((bits >> 16) >= T) {
        unsigned pos = atomicAdd(&cnt[0], 1u);
        if (pos < CAND_MAX)
            cand[pos] = ((unsigned long long)bits << 32) |
                        (unsigned long long)(0xFFFFFFFFu - (unsigned)f);
    }
}

// ---------------- K5: one-block bitonic sort (desc) in LDS, emit top 6000 ----------------
__global__ __launch_bounds__(1024) void sort_emit_kernel(const unsigned long long* __restrict__ cand,
                                                         const unsigned* __restrict__ cnt,
                                                         float* __restrict__ scores_top,
                                                         unsigned* __restrict__ flat_top) {
    __shared__ unsigned long long keys[CAND_MAX];     // 64 KB of the 320 KB LDS
    int tid = threadIdx.x;
    unsigned n = cnt[0];
    if (n > CAND_MAX) n = CAND_MAX;
    for (int i = tid; i < CAND_MAX; i += 1024)
        keys[i] = (i < (int)n) ? cand[i] : 0ULL;      // 0 sinks to the tail
    __syncthreads();
    for (unsigned k = 2; k <= CAND_MAX; k <<= 1) {
        for (unsigned j = k >> 1; j > 0; j >>= 1) {
            for (unsigned i = tid; i < CAND_MAX; i += 1024) {
                unsigned ixj = i ^ j;
                if (ixj > i) {
                    unsigned long long a = keys[i], b = keys[ixj];
                    bool desc = ((i & k) == 0);
                    if (desc ? (a < b) : (a > b)) { keys[i] = b; keys[ixj] = a; }
                }
            }
            __syncthreads();
        }
    }
    for (int r = tid; r < TOPK; r += 1024) {
        unsigned long long key = keys[r];
        scores_top[r] = __uint_as_float((unsigned)(key >> 32));
        flat_top[r]   = 0xFFFFFFFFu - (unsigned)key;
    }
}

// ---------------- K6: decode rois, group-shift, areas, write det/cls/b ----------------
__global__ void decode_kernel(const float* __restrict__ deltas,
                              const float* __restrict__ anchors,
                              const unsigned* __restrict__ flat_top,
                              const float* __restrict__ scores_top,
                              float4* __restrict__ sb,
                              float* __restrict__ areas,
                              float* __restrict__ out) {
    int r = blockIdx.x * blockDim.x + threadIdx.x;
    if (r >= PAD_K) return;
    if (r >= TOPK) {                                  // far-away pad box, never overlaps
        sb[r] = make_float4(-1.0e9f, -1.0e9f, -1.0e9f, -1.0e9f);
        areas[r] = 1.0f;
        return;
    }
    unsigned f = flat_top[r];
    unsigned prop = f >> 1;
    int cls = (int)(f & 1u) + 1;
    int bix = (int)(prop / N_ANCHORS);
    unsigned aix = prop % N_ANCHORS;

    const float inv = 1.0f / SCALE_F;
    float ay1 = anchors[aix * 4 + 0] * inv;
    float ax1 = anchors[aix * 4 + 1] * inv;
    float ay2 = anchors[aix * 4 + 2] * inv;
    float ax2 = anchors[aix * 4 + 3] * inv;
    float d0 = deltas[prop * 4 + 0] * 0.1f;
    float d1 = deltas[prop * 4 + 1] * 0.1f;
    float d2 = deltas[prop * 4 + 2] * 0.2f;
    float d3 = deltas[prop * 4 + 3] * 0.2f;

    float h = ay2 - ay1, w = ax2 - ax1;
    float cy = ay1 + 0.5f * h + d0 * h;
    float cx = ax1 + 0.5f * w + d1 * w;
    h *= expf(d2);
    w *= expf(d3);
    float y1 = (cy - 0.5f * h) * SCALE_F;
    float x1 = (cx - 0.5f * w) * SCALE_F;
    float y2 = y1 + h * SCALE_F;
    float x2 = x1 + w * SCALE_F;
    y1 = fminf(fmaxf(y1, 0.0f), SCALE_F);
    x1 = fminf(fmaxf(x1, 0.0f), SCALE_F);
    y2 = fminf(fmaxf(y2, 0.0f), SCALE_F);
    x2 = fminf(fmaxf(x2, 0.0f), SCALE_F);

    float grp = (float)(bix * (N_FG + 1) + cls) * GROUP_OFF;
    sb[r] = make_float4(y1 + grp, x1 + grp, y2 + grp, x2 + grp);
    areas[r] = (y2 - y1 + 1.0f) * (x2 - x1 + 1.0f);   // shift cancels in area

    out[OUT_DET + r * 6 + 0] = y1;
    out[OUT_DET + r * 6 + 1] = x1;
    out[OUT_DET + r * 6 + 2] = y2;
    out[OUT_DET + r * 6 + 3] = x2;
    out[OUT_DET + r * 6 + 4] = scores_top[r];
    out[OUT_CLS + r] = (float)cls;
    out[OUT_B + r] = (float)bix;
}

// ---------------- K7: NMS suppression-mask tiles (16 rows x 32 cols) ----------------
// Denominator base areas[i]+areas[j] per 16x16 tile is a rank-2 outer sum,
// computed with one V_WMMA_F32_16X16X4_F32:  A row = [area_i, 1, 0, 0],
// B rows = [1...; area_j...; 0; 0]  ->  D[i][j] = area_i + area_j.
// IoU test: inter/denom >= 0.5  <=>  inter >= 0.5*denom   (denom > 0 always;
// 0.5*denom is exact in FP) -- avoids the IEEE-division expansion.
__global__ __launch_bounds__(32) void nms_mask_kernel(const float4* __restrict__ sb,
                                                      const float* __restrict__ areas,
                                                      unsigned* __restrict__ mask) {
    int cb = blockIdx.x;   // column block of 32
    int rb = blockIdx.y;   // row block of 16
    if ((cb + 1) * 32 <= rb * 16) return;   // only j > i region needed (uniform)

    int lane = threadIdx.x;      // wave32: one wave per block
    int l16 = lane & 15;
    bool hi = lane >= 16;
    int row0 = rb * 16, col0 = cb * 32;

    __shared__ float4 cbox[32]; __shared__ float carea[32];
    __shared__ float4 rbox[16]; __shared__ float rarea[16];
    cbox[lane] = sb[col0 + lane];
    carea[lane] = areas[col0 + lane];
    if (lane < 16) {
        rbox[lane] = sb[row0 + lane];
        rarea[lane] = areas[row0 + lane];
    }
    __syncthreads();            // EXEC back to all-1s before WMMA

    // Hoist this lane's 8 row boxes (rows 0-7 or 8-15) into registers once.
    int rbase = hi ? 8 : 0;
    float4 rr_[8];
#pragma unroll
    for (int r = 0; r < 8; ++r) rr_[r] = rbox[rbase + r];

    // A matrix (16x4 f32): lanes 0-15 carry K=0..1, lanes 16-31 carry K=2..3 (zero)
    v2f av;
    av.x = hi ? 0.0f : rarea[l16];   // A[m][0] = area_row[m]
    av.y = hi ? 0.0f : 1.0f;         // A[m][1] = 1

    unsigned myword = 0u;            // mask word this lane will store (lane < 16)

    for (int g = 0; g < 2; ++g) {    // two 16-column halves of the 32-col tile
        int colL = l16 + g * 16;
        // B matrix (4x16 f32): lanes 0-15 carry K=0..1 rows, lanes 16-31 K=2..3 (zero)
        v2f bv;
        bv.x = hi ? 0.0f : 1.0f;          // B[0][n] = 1
        bv.y = hi ? 0.0f : carea[colL];   // B[1][n] = area_col[n]

        float dsum_s[8];
#ifdef HAVE_WMMA_F32_4
        v8f cacc = {};
        v8f dmat = __builtin_amdgcn_wmma_f32_16x16x4_f32(
            /*neg_a=*/false, av, /*neg_b=*/false, bv,
            /*c_mod=*/(short)0, cacc, /*reuse_a=*/false, /*reuse_b=*/false);
#pragma unroll
        for (int r = 0; r < 8; ++r) dsum_s[r] = dmat[r];
#else
#pragma unroll
        for (int r = 0; r < 8; ++r)
            dsum_s[r] = rarea[rbase + r] + carea[colL];
#endif
        float4 cc = cbox[colL];
#pragma unroll
        for (int r = 0; r < 8; ++r) {
            // D layout: lanes 0-15 reg r = (row r, col l16); lanes 16-31 = (row r+8, col l16)
            float4 rr = rr_[r];
            float ihh = fminf(rr.z, cc.z) - fmaxf(rr.x, cc.x) + 1.0f;
            float iww = fminf(rr.w, cc.w) - fmaxf(rr.y, cc.y) + 1.0f;
            float inter = fmaxf(ihh, 0.0f) * fmaxf(iww, 0.0f);
            float denom = dsum_s[r] - inter;
            bool cond = inter >= NMS_THRESH * denom;   // == inter/denom >= 0.5
            unsigned bal = ballot32(cond);   // low 16 bits: row r, high 16: row r+8
            // ballot is wave-uniform: lane r keeps the low half, lane r+8 the high half
            myword |= (lane == r)     ? ((bal & 0xFFFFu) << (g * 16)) : 0u;
            myword |= (lane == r + 8) ? ((bal >> 16)     << (g * 16)) : 0u;
        }
    }
    if (lane < 16)
        mask[(size_t)(row0 + lane) * COL_WORDS + cb] = myword;
}

// ---------------- K8: serial greedy reduce over the bitmask ----------------
__global__ __launch_bounds__(256) void nms_reduce_kernel(const unsigned* __restrict__ mask,
                                                         float* __restrict__ out) {
    __shared__ unsigned remv[COL_WORDS];
    int tid = threadIdx.x;
    for (int i = tid; i < COL_WORDS; i += 256) remv[i] = 0u;
    __syncthreads();
    for (int i = 0; i < TOPK; ++i) {
        int word = i >> 5, bit = i & 31;
        bool kept = ((remv[word] >> bit) & 1u) == 0u;   // uniform across block
        if (kept) {
            const unsigned* mrow = mask + (size_t)i * COL_WORDS;
            for (int c = word + tid; c < COL_WORDS; c += 256) {
                unsigned m = mrow[c];
                if (c == word) m &= (bit == 31) ? 0u : (~0u << (bit + 1));  // only j > i
                remv[c] |= m;
            }
        }
        if (tid == 0) out[OUT_DET + i * 6 + 5] = kept ? 1.0f : 0.0f;
        __syncthreads();
    }
}

// ---------------- host launch ----------------
extern "C" void kernel_launch(void* const* d_in, const int* in_sizes, int n_in,
                              void* d_out, int out_size, void* d_ws, size_t ws_size,
                              hipStream_t stream) {
    const float* probs   = (const float*)d_in[0];   // (522240, 3)
    const float* deltas  = (const float*)d_in[1];   // (522240, 4)
    const float* anchors = (const float*)d_in[2];   // (65280, 4)
    // d_in[3] = batch_ixs: unused (b == prop / N_ANCHORS by construction)
    float* out = (float*)d_out;

    char* ws = (char*)d_ws;
    unsigned*           hist       = (unsigned*)(ws + HIST_OFF);
    unsigned*           cnt        = (unsigned*)(ws + CNT_OFF);
    unsigned long long* cand       = (unsigned long long*)(ws + CAND_OFF);
    float*              scores_top = (float*)(ws + SCORES_OFF);
    unsigned*           flat_top   = (unsigned*)(ws + FLAT_OFF);
    float4*             sb         = (float4*)(ws + SB_OFF);
    float*              areas      = (float*)(ws + AREAS_OFF);
    unsigned*           mask       = (unsigned*)(ws + MASK_OFF);

    init_kernel<<<(NBUCKETS + 8 + 255) / 256, 256, 0, stream>>>(hist);
    hist_kernel<<<(N_FLAT + 255) / 256, 256, 0, stream>>>(probs, hist);
    thresh_kernel<<<1, 256, 0, stream>>>(hist, cnt);
    compact_kernel<<<(N_FLAT + 255) / 256, 256, 0, stream>>>(probs, cnt, cand);
    sort_emit_kernel<<<1, 1024, 0, stream>>>(cand, cnt, scores_top, flat_top);
    decode_kernel<<<(PAD_K + 255) / 256, 256, 0, stream>>>(deltas, anchors, flat_top,
                                                           scores_top, sb, areas, out);
    nms_mask_kernel<<<dim3(COL_BLOCKS, ROW_BLOCKS), 32, 0, stream>>>(sb, areas, mask);
    nms_reduce_kernel<<<1, 256, 0, stream>>>(mask, out);
}